// distribution2_3393024163972
// MI455X (gfx1250) — compile-verified
//
#include <hip/hip_runtime.h>

#define B_ 8
#define N_ 2048
#define M_ 2048
#define M1 2049        // N_+1 == M_+1
#define GAMMA_ 0.5f
#define CCHUNK 4
#define CROWS 513      // ceil(2049/4)

#define ASYNC_PRELOAD 1

#define NEG_INF (-__builtin_inff())

// ---------------------------------------------------------------------------
// Kernel 1: per batch, sort keys K[j] = gt1e[j]*M + j ascending (== stable
// argsort of gt1e), store D_sorted, and gather posvals[k] = s1.ravel()[D_sorted[k]]
// ---------------------------------------------------------------------------
__global__ __launch_bounds__(1024)
void sort_kernel(const float* __restrict__ scores, const int* __restrict__ gt1,
                 int* __restrict__ dsorted, float* __restrict__ posvals)
{
  __shared__ int keys[M_];
  const int b = blockIdx.x;
  const int t = threadIdx.x;
  for (int j = t; j < M_; j += 1024) {
    int g = gt1[b * M_ + j];
    if (g < 0) g = N_;
    keys[j] = g * M_ + j;             // unique composite key, < 2^23
  }
  __syncthreads();
  // bitonic sort of 2048 elements, 1024 threads (one compare-exchange each)
  for (int k = 2; k <= M_; k <<= 1) {
    for (int jj = k >> 1; jj > 0; jj >>= 1) {
      int i = ((t & ~(jj - 1)) << 1) | (t & (jj - 1));
      int partner = i | jj;
      bool up = ((i & k) == 0);
      int a = keys[i], c = keys[partner];
      if ((a > c) == up) { keys[i] = c; keys[partner] = a; }
      __syncthreads();
    }
  }
  const float* sb = scores + (size_t)b * M1 * M1;
  for (int j = t; j < M_; j += 1024) {
    int key = keys[j];
    dsorted[b * M_ + j] = key;
    int row = key >> 11;              // / M_
    int col = key & (M_ - 1);         // % M_
    posvals[b * M_ + j] = sb[(size_t)row * M1 + col];   // s1 row stride is M1
  }
}

// ---------------------------------------------------------------------------
// Kernel 2: column-direction partial top-2 (+argmax row) over a row chunk.
// One thread per column; loads coalesced across lanes; prefetch ahead.
// ---------------------------------------------------------------------------
__global__ __launch_bounds__(256)
void coltop2_partial(const float* __restrict__ scores,
                     float* __restrict__ pv1, float* __restrict__ pv2,
                     int* __restrict__ pi1)
{
  const int b = blockIdx.y, chunk = blockIdx.z;
  const int j = blockIdx.x * 256 + threadIdx.x;
  const int ibeg = chunk * CROWS;
  const int iend = (ibeg + CROWS < M1) ? (ibeg + CROWS) : M1;
  const float* p = scores + (size_t)b * M1 * M1 + j;
  float v1 = NEG_INF, v2 = NEG_INF;
  int i1 = -1;
  for (int i = ibeg; i < iend; ++i) {
    if (i + 24 < iend) __builtin_prefetch(p + (size_t)(i + 24) * M1, 0, 1);
    float v = p[(size_t)i * M1];
    if (v > v1) { v2 = v1; v1 = v; i1 = i; }
    else if (v > v2) { v2 = v; }
  }
  const size_t idx = ((size_t)b * CCHUNK + chunk) * M_ + j;
  pv1[idx] = v1; pv2[idx] = v2; pi1[idx] = i1;
}

// ---------------------------------------------------------------------------
// Kernel 3: merge chunk partials -> hard_neg1 per column, pair with posvals[k]
// (positional pairing, as in the reference), accumulate hard-margin sum.
// ---------------------------------------------------------------------------
__global__ __launch_bounds__(256)
void colmerge_kernel(const float* __restrict__ pv1, const float* __restrict__ pv2,
                     const int* __restrict__ pi1, const int* __restrict__ gt1,
                     const float* __restrict__ posvals, double* __restrict__ accum)
{
  const int b = blockIdx.y;
  const int j = blockIdx.x * 256 + threadIdx.x;
  float v1 = NEG_INF, v2 = NEG_INF; int i1 = 0x7fffffff;
  for (int c = 0; c < CCHUNK; ++c) {
    size_t idx = ((size_t)b * CCHUNK + c) * M_ + j;
    float a1 = pv1[idx]; float a2 = pv2[idx]; int ai = pi1[idx];
    if (a1 > v1 || (a1 == v1 && ai < i1)) { v2 = fmaxf(v1, a2); v1 = a1; i1 = ai; }
    else                                  { v2 = fmaxf(v2, a1); }
  }
  int g = gt1[b * M_ + j]; if (g < 0) g = N_;
  float hn = (i1 == g) ? v2 : v1;
  float h  = posvals[b * M_ + j] - hn;
  float contrib = fmaxf(h + GAMMA_, 0.0f);

  __shared__ float red[256];
  red[threadIdx.x] = contrib; __syncthreads();
  for (int off = 128; off > 0; off >>= 1) {
    if (threadIdx.x < off) red[threadIdx.x] += red[threadIdx.x + off];
    __syncthreads();
  }
  if (threadIdx.x == 0) atomicAdd(&accum[2], (double)red[0]);
}

// ---------------------------------------------------------------------------
// Kernel 4: row-direction pass: per row (b,i): S=Σs, Q=Σs², top2+argmax over
// all M1 columns.  margin1 row sums collapse algebraically:
//   Σ(neg-pos) = S - M1*pos ;  Σ(neg-pos)² = Q - 2*pos*S + M1*pos²
// ---------------------------------------------------------------------------
__global__ __launch_bounds__(256)
void row_kernel(const float* __restrict__ scores, const int* __restrict__ gt0,
                double* __restrict__ accum)
{
  const int row = blockIdx.x;
  const int b = row >> 11;            // / N_
  const int i = row & (N_ - 1);       // % N_
  const float* p = scores + (size_t)b * M1 * M1 + (size_t)i * M1;
  float s = 0.f, q = 0.f, v1 = NEG_INF, v2 = NEG_INF; int i1 = 0x7fffffff;
  for (int c = threadIdx.x; c < M1; c += 256) {
    float v = p[c];
    s += v; q = fmaf(v, v, q);
    if (v > v1) { v2 = v1; v1 = v; i1 = c; }
    else if (v > v2) v2 = v;
  }
  __shared__ float rs[256], rq[256], rv1[256], rv2[256];
  __shared__ int   ri1[256];
  const int t = threadIdx.x;
  rs[t] = s; rq[t] = q; rv1[t] = v1; rv2[t] = v2; ri1[t] = i1;
  __syncthreads();
  for (int off = 128; off > 0; off >>= 1) {
    if (t < off) {
      rs[t] += rs[t + off]; rq[t] += rq[t + off];
      float b1 = rv1[t + off], b2 = rv2[t + off]; int bi = ri1[t + off];
      float a1 = rv1[t],       a2 = rv2[t];       int ai = ri1[t];
      if (b1 > a1 || (b1 == a1 && bi < ai)) { rv1[t] = b1; ri1[t] = bi; rv2[t] = fmaxf(a1, b2); }
      else                                  { rv2[t] = fmaxf(a2, b1); }
    }
    __syncthreads();
  }
  if (t == 0) {
    int g = gt0[b * N_ + i]; if (g < 0) g = M_;
    float pos = p[g];
    float S = rs[0], Q = rq[0];
    float msum = S - (float)M1 * pos;
    float msq  = Q - 2.f * pos * S + (float)M1 * pos * pos;
    float hn   = (ri1[0] == g) ? rv2[0] : rv1[0];
    float hard = fmaxf(pos - hn + GAMMA_, 0.f);
    atomicAdd(&accum[0], (double)msum);
    atomicAdd(&accum[1], (double)msq);
    atomicAdd(&accum[2], (double)hard);
  }
}

// ---------------------------------------------------------------------------
// Kernel 5: margin2 pass.  For kept element r=i*M+j of s1: rank
// q = r - (#deleted keys < r); paired positive = posvals[q % M].
// D_sorted + posvals staged to LDS via CDNA5 async global->LDS copies.
// Each thread visits r in strictly increasing order, so the lower_bound is
// maintained INCREMENTALLY (one binary search at entry, then a register
// compare per element; LDS touched only when crossing a deleted key).
// ---------------------------------------------------------------------------
__global__ __launch_bounds__(256)
void margin2_kernel(const float* __restrict__ scores, const int* __restrict__ dsorted,
                    const float* __restrict__ posvals, double* __restrict__ accum)
{
  __shared__ __align__(16) int   sd[M_];
  __shared__ __align__(16) float sp[M_];
  const int b = blockIdx.y;
#if ASYNC_PRELOAD
  {
    const int*   gsd = dsorted + b * M_;
    const float* gsp = posvals + b * M_;
    for (int t4 = threadIdx.x * 4; t4 < M_; t4 += 256 * 4) {
      unsigned d0 = (unsigned)(unsigned long long)(uintptr_t)&sd[t4];
      const int* s0 = gsd + t4;
      asm volatile("global_load_async_to_lds_b128 %0, %1, off" :: "v"(d0), "v"(s0) : "memory");
      unsigned d1 = (unsigned)(unsigned long long)(uintptr_t)&sp[t4];
      const float* s1 = gsp + t4;
      asm volatile("global_load_async_to_lds_b128 %0, %1, off" :: "v"(d1), "v"(s1) : "memory");
    }
    asm volatile("s_wait_asynccnt 0x0" ::: "memory");
  }
  __syncthreads();
#else
  for (int t = threadIdx.x; t < M_; t += 256) {
    sd[t] = dsorted[b * M_ + t]; sp[t] = posvals[b * M_ + t];
  }
  __syncthreads();
#endif
  const int row0 = blockIdx.x * 8;
  // one binary search for this thread's first r; then advance incrementally
  int lo;
  {
    int rfirst = row0 * M_ + (int)threadIdx.x;
    int l = 0, h = M_;
    while (l < h) { int mid = (l + h) >> 1; if (sd[mid] < rfirst) l = mid + 1; else h = mid; }
    lo = l;
  }
  int cur = (lo < M_) ? sd[lo] : 0x7fffffff;   // cached sd[lo]

  float lsum = 0.f, lsq = 0.f;
  for (int rr = 0; rr < 8; ++rr) {
    int i = row0 + rr;
    if (i >= M1) break;
    const float* p = scores + (size_t)b * M1 * M1 + (size_t)i * M1;
    for (int j = threadIdx.x; j < M_; j += 256) {
      int r = i * M_ + j;                 // strictly increasing per thread
      while (cur < r) { ++lo; cur = (lo < M_) ? sd[lo] : 0x7fffffff; }
      if (cur == r) continue;             // this element is the deleted positive
      int qq = r - lo;
      float diff = p[j] - sp[qq & (M_ - 1)];
      lsum += diff; lsq = fmaf(diff, diff, lsq);
    }
  }
  __shared__ float r1[256], r2[256];
  const int t = threadIdx.x;
  r1[t] = lsum; r2[t] = lsq;
  __syncthreads();
  for (int off = 128; off > 0; off >>= 1) {
    if (t < off) { r1[t] += r1[t + off]; r2[t] += r2[t + off]; }
    __syncthreads();
  }
  if (t == 0) { atomicAdd(&accum[0], (double)r1[0]); atomicAdd(&accum[1], (double)r2[0]); }
}

// ---------------------------------------------------------------------------
// Kernel 6: finalize scalar loss.
// ---------------------------------------------------------------------------
__global__ void finalize_kernel(const double* __restrict__ accum, float* __restrict__ out)
{
  const double cm = 2.0 * (double)B_ * (double)N_ * (double)M_;  // margin count
  const double ch = 2.0 * (double)B_ * (double)N_;               // hard-margin count
  double msum = accum[0], msq = accum[1], hsum = accum[2];
  double mean = msum / cm;
  double var  = (msq - msum * msum / cm) / (cm - 1.0);           // var(-m) == var(m)
  double loss = hsum / ch + exp(mean) + log(var + 1.0);
  out[0] = (float)loss;
}

// ---------------------------------------------------------------------------
extern "C" void kernel_launch(void* const* d_in, const int* in_sizes, int n_in,
                              void* d_out, int out_size, void* d_ws, size_t ws_size,
                              hipStream_t stream)
{
  (void)in_sizes; (void)n_in; (void)out_size; (void)ws_size;
  const float* scores = (const float*)d_in[0];
  const int*   gt0    = (const int*)d_in[1];
  const int*   gt1    = (const int*)d_in[2];
  float* out = (float*)d_out;

  char* ws = (char*)d_ws;
  double* accum   = (double*)ws;                              // 3 doubles
  int*    dsorted = (int*)(ws + 64);                          // B*M ints (16B aligned)
  float*  posvals = (float*)(ws + 64 + (size_t)B_ * M_ * 4);  // B*M floats
  float*  pv1     = (float*)(ws + 64 + 2 * (size_t)B_ * M_ * 4);
  float*  pv2     = pv1 + (size_t)B_ * CCHUNK * M_;
  int*    pi1     = (int*)(pv2 + (size_t)B_ * CCHUNK * M_);

  hipMemsetAsync(accum, 0, 3 * sizeof(double), stream);
  sort_kernel   <<<B_, 1024, 0, stream>>>(scores, gt1, dsorted, posvals);
  coltop2_partial<<<dim3(M_ / 256, B_, CCHUNK), 256, 0, stream>>>(scores, pv1, pv2, pi1);
  colmerge_kernel<<<dim3(M_ / 256, B_), 256, 0, stream>>>(pv1, pv2, pi1, gt1, posvals, accum);
  row_kernel    <<<B_ * N_, 256, 0, stream>>>(scores, gt0, accum);
  margin2_kernel<<<dim3((M1 + 7) / 8, B_), 256, 0, stream>>>(scores, dsorted, posvals, accum);
  finalize_kernel<<<1, 1, 0, stream>>>(accum, out);
}